// StateTransition_45707041964068
// MI455X (gfx1250) — compile-verified
//
#include <hip/hip_runtime.h>

// Lorenz RK4 step — bandwidth-bound streaming kernel for MI455X (gfx1250).
// 8M states x 3 f32. 192 MB traffic/call -> ~8.2 us floor at 23.3 TB/s HBM.
// Strategy: 4 states per thread via 3x b128 loads / 3x b128 NT stores,
// RT loads (input stays resident in 192MB L2 across graph replays),
// NT stores (write-once output must not evict the input working set).

#define DT_F     0.01f
#define SIGMA_F  10.0f
#define RHO_F    28.0f
#define BETA_F   (8.0f / 3.0f)

typedef __attribute__((ext_vector_type(4))) float f4;

__device__ __forceinline__ void lorenz_f(float x, float y, float z,
                                         float& dx, float& dy, float& dz) {
    dx = SIGMA_F * (y - x);
    dy = fmaf(x, RHO_F - z, -y);   // x*(RHO - z) - y
    dz = fmaf(x, y, -BETA_F * z);  // x*y - BETA*z
}

__device__ __forceinline__ void rk4_state(float& x, float& y, float& z) {
    const float h2 = 0.5f * DT_F;
    const float h6 = DT_F / 6.0f;
    float k1x, k1y, k1z, k2x, k2y, k2z, k3x, k3y, k3z, k4x, k4y, k4z;

    lorenz_f(x, y, z, k1x, k1y, k1z);
    lorenz_f(fmaf(h2, k1x, x), fmaf(h2, k1y, y), fmaf(h2, k1z, z),
             k2x, k2y, k2z);
    lorenz_f(fmaf(h2, k2x, x), fmaf(h2, k2y, y), fmaf(h2, k2z, z),
             k3x, k3y, k3z);
    lorenz_f(fmaf(DT_F, k3x, x), fmaf(DT_F, k3y, y), fmaf(DT_F, k3z, z),
             k4x, k4y, k4z);

    const float sx = k1x + 2.0f * (k2x + k3x) + k4x;
    const float sy = k1y + 2.0f * (k2y + k3y) + k4y;
    const float sz = k1z + 2.0f * (k2z + k3z) + k4z;
    x = fmaf(h6, sx, x);
    y = fmaf(h6, sy, y);
    z = fmaf(h6, sz, z);
}

__global__ void __launch_bounds__(256)
lorenz_rk4_kernel(const float* __restrict__ xin,
                  float* __restrict__ xout,
                  int n_states) {
    const long long t = (long long)blockIdx.x * blockDim.x + threadIdx.x;
    const long long s0 = t * 4;  // first state owned by this thread

    if (s0 + 4 <= (long long)n_states) {
        // 4 states = 12 floats = 3 x float4, contiguous 48B per thread.
        const f4* __restrict__ in4 = reinterpret_cast<const f4*>(xin) + t * 3;
        f4 a = in4[0];   // s0:(x,y,z)  s1:(x)
        f4 b = in4[1];   // s1:(y,z)    s2:(x,y)
        f4 c = in4[2];   // s2:(z)      s3:(x,y,z)

        // Unpack: vector elements are not reference-bindable lvalues.
        float s0x = a.x, s0y = a.y, s0z = a.z;
        float s1x = a.w, s1y = b.x, s1z = b.y;
        float s2x = b.z, s2y = b.w, s2z = c.x;
        float s3x = c.y, s3y = c.z, s3z = c.w;

        rk4_state(s0x, s0y, s0z);
        rk4_state(s1x, s1y, s1z);
        rk4_state(s2x, s2y, s2z);
        rk4_state(s3x, s3y, s3z);

        a.x = s0x; a.y = s0y; a.z = s0z; a.w = s1x;
        b.x = s1y; b.y = s1z; b.z = s2x; b.w = s2y;
        c.x = s2z; c.y = s3x; c.z = s3y; c.w = s3z;

        f4* __restrict__ o4 = reinterpret_cast<f4*>(xout) + t * 3;
        // Non-temporal: output is write-once, keep L2 for the input stream.
        __builtin_nontemporal_store(a, o4 + 0);
        __builtin_nontemporal_store(b, o4 + 1);
        __builtin_nontemporal_store(c, o4 + 2);
    } else if (s0 < (long long)n_states) {
        // Scalar tail (unused for N = 8,000,000 but kept for generality).
        for (long long s = s0; s < (long long)n_states; ++s) {
            float x = xin[3 * s + 0];
            float y = xin[3 * s + 1];
            float z = xin[3 * s + 2];
            rk4_state(x, y, z);
            xout[3 * s + 0] = x;
            xout[3 * s + 1] = y;
            xout[3 * s + 2] = z;
        }
    }
}

extern "C" void kernel_launch(void* const* d_in, const int* in_sizes, int n_in,
                              void* d_out, int out_size, void* d_ws, size_t ws_size,
                              hipStream_t stream) {
    (void)n_in; (void)d_ws; (void)ws_size; (void)out_size;
    const float* x = (const float*)d_in[0];
    float* out = (float*)d_out;

    const int n_states = in_sizes[0] / 3;          // [N, 3] flattened
    const int threads_needed = (n_states + 3) / 4; // 4 states per thread
    const int block = 256;                         // 8 waves of 32 on CDNA5
    const int grid = (threads_needed + block - 1) / block;

    lorenz_rk4_kernel<<<grid, block, 0, stream>>>(x, out, n_states);
}